// SPH3D_472446403139
// MI455X (gfx1250) — compile-verified
//
#include <hip/hip_runtime.h>
#include <hip/hip_bf16.h>
#include <math.h>

// ---------------------------------------------------------------------------
// SPH3D-GCN forward pass for MI455X (gfx1250, wave32, WMMA).
// Dense matmuls: LDS-staged tiled V_WMMA_F32_16X16X4_F32 GEMM with
// DOUBLE-BUFFERED ASYNC staging (GLOBAL_LOAD_ASYNC_TO_LDS_B128 + ASYNCcnt):
// the next K-tile's A/B panels are DMA'd straight into LDS while the current
// tile's 8 chained WMMAs execute. All GEMM operands padded to tile multiples
// (M%64, N%64, K%16), so there is zero bounds logic anywhere in the GEMM.
// ---------------------------------------------------------------------------

#define PI_F 3.14159265358979323846f
#define KNN_K 32

typedef float v2f __attribute__((ext_vector_type(2)));
typedef float v8f __attribute__((ext_vector_type(8)));

static inline int cdiv(int a, int b) { return (a + b - 1) / b; }

// ---------------------------------------------------------------------------
// Tiled fp32 WMMA GEMM: C[M,N] = A[M,K] * B[K,N], row-major.
// REQUIRES: M % 64 == 0, N % 64 == 0, K % 16 == 0, lda % 16 == 0 (floats),
//           A/B/C 16B-aligned. Caller guarantees via padding.
//  - Block: 256 threads (8 waves). C tile: 64 (M) x 64 (N).
//  - Wave w -> (wm = w&3) 16-row band, (wn = w>>2) 32-col strip: two 16x16
//    WMMA subtiles sharing the A fragment -> 8 WMMAs per K tile per wave.
//  - Staging: per K-tile each thread issues 2x global_load_async_to_lds_b128
//    (A panel 64x16, B panel 16x64) into the OTHER LDS buffer, overlapping
//    DMA with WMMA; one s_wait_asynccnt 0 + barrier per tile.
//  - WMMA VGPR layouts (ISA 7.12.2): A VGPR0/1 = K{0|2}/{1|3} per lane half;
//    B row per VGPR (K{0|2}/{1|3}), N = lane&15; C/D VGPR v -> row v+8*half.
//  - EXEC is all-ones at every WMMA (only uniform control flow).
// ---------------------------------------------------------------------------
#define GEMM_MT 64
#define GEMM_NT 64
#define GEMM_KT 16
#define SAP 20   // sA row stride (floats): %4 -> 16B-aligned b128 DMA;
                 // 20*r mod 64 distinct for r=0..15 -> conflict-free b64 reads
#define SBP 68   // sB row stride (floats): %4 -> 16B-aligned b128 DMA

__global__ __launch_bounds__(256)
void k_gemm_wmma(const float* __restrict__ A, const float* __restrict__ B,
                 float* __restrict__ C, int M, int N, int K,
                 int lda, int ldb, int ldc) {
    __shared__ float sA[2][GEMM_MT][SAP];
    __shared__ float sB[2][GEMM_KT][SBP];

    const int tid  = threadIdx.x;
    const int lane = tid & 31;
    const int wave = tid >> 5;
    const int r    = lane & 15;
    const int h    = lane >> 4;            // lane-half select
    const int wm   = wave & 3;             // M subtile (0..3)
    const int wn   = wave >> 2;            // N strip (0..1)
    const int mBlk = blockIdx.y * GEMM_MT;
    const int nBlk = blockIdx.x * GEMM_NT;
    const int aRow  = wm * 16 + r;         // row in sA for this lane
    const int bCol0 = wn * 32 + r;         // cols in sB for the two subtiles
    const int bCol1 = bCol0 + 16;

    // staging coordinates (one 16B async DMA per thread per panel)
    const int aSRow = tid >> 2;            // 0..63
    const int aSK   = (tid & 3) * 4;       // 0,4,8,12
    const int bSRow = tid >> 4;            // 0..15
    const int bSCol = (tid & 15) * 4;      // 0..60

    const float* aSrc = &A[(size_t)(mBlk + aSRow) * lda + aSK];
    const float* bSrc = &B[(size_t)bSRow * ldb + nBlk + bSCol];
    // LDS byte offsets for the async DMA destinations (generic shared pointer
    // truncates to the 32-bit LDS address per ISA 10.2).
    const unsigned ldsA0 = (unsigned)(unsigned long long)(void*)&sA[0][aSRow][aSK];
    const unsigned ldsA1 = (unsigned)(unsigned long long)(void*)&sA[1][aSRow][aSK];
    const unsigned ldsB0 = (unsigned)(unsigned long long)(void*)&sB[0][bSRow][bSCol];
    const unsigned ldsB1 = (unsigned)(unsigned long long)(void*)&sB[1][bSRow][bSCol];

    #define STAGE_ASYNC(buf, kt)                                              \
        do {                                                                  \
            const float* ga = aSrc + (kt);                                    \
            const float* gb = bSrc + (size_t)(kt) * ldb;                      \
            asm volatile("global_load_async_to_lds_b128 %0, %1, off"          \
                         :: "v"((buf) ? ldsA1 : ldsA0), "v"(ga) : "memory");  \
            asm volatile("global_load_async_to_lds_b128 %0, %1, off"          \
                         :: "v"((buf) ? ldsB1 : ldsB0), "v"(gb) : "memory");  \
        } while (0)

    // prologue: DMA first tile into buffer 0
    STAGE_ASYNC(0, 0);
    asm volatile("s_wait_asynccnt 0x0" ::: "memory");
    __syncthreads();

    v8f acc0 = {};
    v8f acc1 = {};
    int cur = 0;
    for (int kt = 0; kt < K; kt += GEMM_KT) {
        // kick off DMA of the next tile into the other buffer
        if (kt + GEMM_KT < K) STAGE_ASYNC(cur ^ 1, kt + GEMM_KT);
        // ---- 4 k-substeps x 2 WMMAs (shared A fragment) on buffer `cur` ----
        #pragma unroll
        for (int ks = 0; ks < GEMM_KT; ks += 4) {
            const int ka = ks + 2 * h;                   // even -> 8B aligned
            v2f a = *(const v2f*)&sA[cur][aRow][ka];     // ds_load_b64
            v2f b0, b1;
            b0[0] = sB[cur][ka][bCol0];
            b0[1] = sB[cur][ka + 1][bCol0];
            b1[0] = sB[cur][ka][bCol1];
            b1[1] = sB[cur][ka + 1][bCol1];
            acc0 = __builtin_amdgcn_wmma_f32_16x16x4_f32(
                false, a, false, b0, (short)0, acc0, false, false);
            acc1 = __builtin_amdgcn_wmma_f32_16x16x4_f32(
                false, a, false, b1, (short)0, acc1, false, false);
        }
        asm volatile("s_wait_asynccnt 0x0" ::: "memory");
        __syncthreads();
        cur ^= 1;
    }
    #undef STAGE_ASYNC

    // ---- store the two 16x16 subtiles (unconditional; C is padded) ----
    const int col0 = nBlk + wn * 32 + r;
    const int col1 = col0 + 16;
    #pragma unroll
    for (int v = 0; v < 8; ++v) {
        const int row = mBlk + wm * 16 + v + 8 * h;
        C[(size_t)row * ldc + col0] = acc0[v];
        C[(size_t)row * ldc + col1] = acc1[v];
    }
}

// Zero-fill helper (pad regions refreshed every call -> deterministic).
__global__ void k_zero(float* __restrict__ p, int total) {
    const int t = blockIdx.x * blockDim.x + threadIdx.x;
    if (t < total) p[t] = 0.0f;
}

// Pad-copy a (K,N) weight into a zero-padded (Kp,Np) workspace copy.
__global__ void k_pad_copy(const float* __restrict__ src, int K, int N,
                           float* __restrict__ dst, int Np, int total) {
    const int t = blockIdx.x * blockDim.x + threadIdx.x;
    if (t >= total) return;
    const int k = t / Np, n = t - k * Np;
    dst[t] = (k < K && n < N) ? src[(size_t)k * N + n] : 0.0f;
}

// ---------------------------------------------------------------------------
// Per-batch centering / unit-sphere scaling; emits xyz (B,N,3), a K-padded
// copy xyzPad (B*N,16) for the fc1 GEMM, and query = mean(xyz).
// pc is (B,3,N).
// ---------------------------------------------------------------------------
__global__ void k_center_scale(const float* __restrict__ pc, float* __restrict__ xyz,
                               float* __restrict__ xyzPad, float* __restrict__ query,
                               int N) {
    __shared__ float red[256];
    const int b = blockIdx.x, t = threadIdx.x;
    const float* p = pc + (size_t)b * 3 * N;

    float mean[3];
    for (int d = 0; d < 3; ++d) {
        float s = 0.0f;
        for (int i = t; i < N; i += 256) s += p[(size_t)d * N + i];
        red[t] = s; __syncthreads();
        for (int o = 128; o > 0; o >>= 1) { if (t < o) red[t] += red[t + o]; __syncthreads(); }
        mean[d] = red[0] / (float)N; __syncthreads();
    }
    float mx = 0.0f;
    for (int i = t; i < N; i += 256) {
        float x = p[i] - mean[0];
        float y = p[(size_t)N + i] - mean[1];
        float z = p[(size_t)2 * N + i] - mean[2];
        mx = fmaxf(mx, x * x + y * y + z * z);
    }
    red[t] = mx; __syncthreads();
    for (int o = 128; o > 0; o >>= 1) { if (t < o) red[t] = fmaxf(red[t], red[t + o]); __syncthreads(); }
    const float inv = 1.0f / sqrtf(red[0]); __syncthreads();

    float qs[3] = {0.f, 0.f, 0.f};
    for (int i = t; i < N; i += 256) {
        float x = (p[i] - mean[0]) * inv;
        float y = (p[(size_t)N + i] - mean[1]) * inv;
        float z = (p[(size_t)2 * N + i] - mean[2]) * inv;
        float* o = xyz + ((size_t)b * N + i) * 3;
        o[0] = x; o[1] = y; o[2] = z;
        float* op = xyzPad + ((size_t)b * N + i) * 16;
        op[0] = x; op[1] = y; op[2] = z;     // cols 3..15 pre-zeroed
        qs[0] += x; qs[1] += y; qs[2] += z;
    }
    for (int d = 0; d < 3; ++d) {
        red[t] = qs[d]; __syncthreads();
        for (int o = 128; o > 0; o >>= 1) { if (t < o) red[t] += red[t + o]; __syncthreads(); }
        if (t == 0) query[b * 3 + d] = red[0] / (float)N;
        __syncthreads();
    }
}

// ---------------------------------------------------------------------------
// kNN graph: per point keep 32 smallest d2 (stable ties -> lower index first,
// matching jax top_k). Candidate lists live in LDS; point tiles staged in LDS;
// next tile prefetched into cache (global_prefetch_b8).
// ---------------------------------------------------------------------------
__global__ void k_build_graph(const float* __restrict__ xyz, int N, float radius,
                              int* __restrict__ idxO, float* __restrict__ dstO,
                              float* __restrict__ validO) {
    const int b = blockIdx.y;
    const int i = blockIdx.x * 64 + threadIdx.x;
    const float* X = xyz + (size_t)b * N * 3;

    __shared__ float sd[64][KNN_K];
    __shared__ int   si[64][KNN_K];
    __shared__ float sx[128][3];

    float qx = 0.f, qy = 0.f, qz = 0.f;
    if (i < N) { qx = X[i * 3]; qy = X[i * 3 + 1]; qz = X[i * 3 + 2]; }
    const float qsq = qx * qx + qy * qy + qz * qz;

    float* bd = sd[threadIdx.x];
    int*   bi = si[threadIdx.x];
    for (int k = 0; k < KNN_K; ++k) { bd[k] = 3.4e38f; bi[k] = i; }

    for (int j0 = 0; j0 < N; j0 += 128) {
        __syncthreads();
        for (int tidx = threadIdx.x; tidx < 128 * 3; tidx += 64) {
            const int jj = j0 + tidx / 3;
            sx[tidx / 3][tidx % 3] = (jj < N) ? X[jj * 3 + tidx % 3] : 0.0f;
        }
        if (j0 + 128 < N)
            __builtin_prefetch(&X[(size_t)(j0 + 128) * 3 + threadIdx.x * 6], 0, 3);
        __syncthreads();
        if (i < N) {
            const int lim = min(128, N - j0);
            for (int tt = 0; tt < lim; ++tt) {
                const float px = sx[tt][0], py = sx[tt][1], pz = sx[tt][2];
                float d2 = qsq + px * px + py * py + pz * pz
                         - 2.0f * (qx * px + qy * py + qz * pz);
                d2 = fmaxf(d2, 0.0f);
                if (d2 < bd[KNN_K - 1]) {          // strict: keep lower-index on tie
                    int k = KNN_K - 1;
                    while (k > 0 && bd[k - 1] > d2) {
                        bd[k] = bd[k - 1]; bi[k] = bi[k - 1]; --k;
                    }
                    bd[k] = d2; bi[k] = j0 + tt;
                }
            }
        }
    }
    if (i < N) {
        for (int k = 0; k < KNN_K; ++k) {
            const float dst = sqrtf(fmaxf(bd[k], 1e-12f));
            const bool  ok  = dst <= radius;
            const size_t o  = ((size_t)b * N + i) * KNN_K + k;
            idxO[o]   = ok ? bi[k] : i;
            dstO[o]   = ok ? dst : 0.0f;
            validO[o] = ok ? 1.0f : 0.0f;
        }
    }
}

// ---------------------------------------------------------------------------
// Spherical kernel binning for a level: kernel (8,2,2), self-bin 32.
// ---------------------------------------------------------------------------
__global__ void k_sph_filt(const float* __restrict__ xyz, const int* __restrict__ idx,
                           const float* __restrict__ dst, int N, float radius,
                           int* __restrict__ filt, int total) {
    const int t = blockIdx.x * blockDim.x + threadIdx.x;
    if (t >= total) return;
    const int b = t / (N * KNN_K);
    const int rem = t - b * N * KNN_K;
    const int m = rem / KNN_K;
    const float* C = xyz + ((size_t)b * N + m) * 3;
    const float* P = xyz + ((size_t)b * N + idx[t]) * 3;
    const float dx = P[0] - C[0], dy = P[1] - C[1], dz = P[2] - C[2];
    const float d  = dst[t];
    const float safe = fmaxf(d, 1e-8f);
    const float theta = atan2f(dy, dx);
    float binA = floorf((theta + PI_F) / (2.0f * PI_F / 8.0f));
    binA = fminf(fmaxf(binA, 0.0f), 7.0f);
    const float phi = asinf(fminf(fmaxf(dz / safe, -1.0f), 1.0f));
    float binP = floorf((phi + PI_F * 0.5f) / (PI_F / 2.0f));
    binP = fminf(fmaxf(binP, 0.0f), 1.0f);
    float binR = floorf(d / (radius / 2.0f));
    binR = fminf(fmaxf(binR, 0.0f), 1.0f);
    int bb = (int)((binR * 2.0f + binP) * 8.0f + binA);
    if (d < 1e-5f) bb = 32;
    filt[t] = bb;
}

// Global stage binning: kernel (8,2,1), radius 100, self-bin 16; also dst calc.
__global__ void k_global_filt(const float* __restrict__ xyz, const float* __restrict__ query,
                              int N, int* __restrict__ filt, int total) {
    const int t = blockIdx.x * blockDim.x + threadIdx.x;
    if (t >= total) return;
    const int b = t / N, j = t - b * N;
    const float* P = xyz + ((size_t)b * N + j) * 3;
    const float dx = P[0] - query[b * 3 + 0];
    const float dy = P[1] - query[b * 3 + 1];
    const float dz = P[2] - query[b * 3 + 2];
    const float d  = sqrtf(dx * dx + dy * dy + dz * dz + 1e-12f);
    const float safe = fmaxf(d, 1e-8f);
    const float theta = atan2f(dy, dx);
    float binA = fminf(fmaxf(floorf((theta + PI_F) / (2.0f * PI_F / 8.0f)), 0.0f), 7.0f);
    const float phi = asinf(fminf(fmaxf(dz / safe, -1.0f), 1.0f));
    float binP = fminf(fmaxf(floorf((phi + PI_F * 0.5f) / (PI_F / 2.0f)), 0.0f), 1.0f);
    int bb = (int)(binP * 8.0f + binA);        // binR == 0 always (radius 100)
    if (d < 1e-5f) bb = 16;
    filt[t] = bb;
}

// ---------------------------------------------------------------------------
// Depthwise neighbor aggregation with padded output stride:
// out[b,m,c] = sum_k net[b,idx,c]*valid*depth[filt,c] for c<C; 0 for C<=c<Cp.
// ---------------------------------------------------------------------------
__global__ void k_depthwise(const float* __restrict__ net, int ldn,
                            const int* __restrict__ idx,
                            const int* __restrict__ filt, const float* __restrict__ valid,
                            const float* __restrict__ depth, int N, int C, int Cp,
                            float* __restrict__ out) {
    const int b = blockIdx.y, m = blockIdx.x;
    __shared__ int   sI[KNN_K];
    __shared__ int   sF[KNN_K];
    __shared__ float sV[KNN_K];
    const size_t base = ((size_t)b * N + m) * KNN_K;
    if (threadIdx.x < KNN_K) {
        sI[threadIdx.x] = idx[base + threadIdx.x];
        sF[threadIdx.x] = filt[base + threadIdx.x];
        sV[threadIdx.x] = valid[base + threadIdx.x];
    }
    __syncthreads();
    for (int c = threadIdx.x; c < Cp; c += blockDim.x) {
        float acc = 0.0f;
        if (c < C) {
            for (int k = 0; k < KNN_K; ++k) {
                acc += net[((size_t)b * N + sI[k]) * ldn + c] * sV[k] * depth[sF[k] * C + c];
            }
        }
        out[((size_t)b * N + m) * Cp + c] = acc;
    }
}

// Global depthwise: d[b,c] = sum_j net[b,j,c] * depth[gfilt[b,j], c]
__global__ void k_global_dw(const float* __restrict__ net, const int* __restrict__ gfilt,
                            const float* __restrict__ depth, int Nf, int C,
                            float* __restrict__ out) {
    const int b = blockIdx.x;
    for (int c = threadIdx.x; c < C; c += blockDim.x) {
        float acc = 0.0f;
        for (int j = 0; j < Nf; ++j) {
            acc += net[((size_t)b * Nf + j) * C + c] * depth[gfilt[b * Nf + j] * C + c];
        }
        out[(size_t)b * C + c] = acc;
    }
}

// ---------------------------------------------------------------------------
// Column stats for training-mode BN: mean + rsqrt(var + 1e-5) per column.
// ---------------------------------------------------------------------------
__global__ void k_colstats(const float* __restrict__ X, int M, int N, int ldx,
                           float* __restrict__ mean, float* __restrict__ inv) {
    __shared__ float s1[256];
    __shared__ float s2[256];
    const int c = blockIdx.x, t = threadIdx.x;
    float a = 0.0f, q = 0.0f;
    for (int r = t; r < M; r += 256) {
        const float v = X[(size_t)r * ldx + c];
        a += v; q += v * v;
    }
    s1[t] = a; s2[t] = q; __syncthreads();
    for (int o = 128; o > 0; o >>= 1) {
        if (t < o) { s1[t] += s1[t + o]; s2[t] += s2[t + o]; }
        __syncthreads();
    }
    if (t == 0) {
        const float mu = s1[0] / (float)M;
        const float var = s2[0] / (float)M - mu * mu;
        mean[c] = mu;
        inv[c]  = rsqrtf(var + 1e-5f);
    }
}

// Fused BN (+optional ReLU), optional strided scatter into a concat buffer.
__global__ void k_bn_act(const float* __restrict__ X, int M, int N, int ldx,
                         const float* __restrict__ mean, const float* __restrict__ inv,
                         const float* __restrict__ gamma, const float* __restrict__ beta,
                         float* __restrict__ out, int ldo, int colOff, int relu) {
    const int t = blockIdx.x * blockDim.x + threadIdx.x;
    if (t >= M * N) return;
    const int r = t / N, c = t - r * N;
    float v = gamma[c] * (X[(size_t)r * ldx + c] - mean[c]) * inv[c] + beta[c];
    if (relu) v = fmaxf(v, 0.0f);
    out[(size_t)r * ldo + colOff + c] = v;
}

// net_c = concat(net, xyz) on channel axis, zero-padded to ldo channels.
__global__ void k_concat(const float* __restrict__ net, int ldn,
                         const float* __restrict__ xyz,
                         int N, int C, float* __restrict__ out, int ldo, int total) {
    const int t = blockIdx.x * blockDim.x + threadIdx.x;
    if (t >= total) return;
    const int bm = t / ldo, c = t - bm * ldo;
    float v = 0.0f;
    if (c < C)          v = net[(size_t)bm * ldn + c];
    else if (c < C + 3) v = xyz[(size_t)bm * 3 + (c - C)];
    out[t] = v;
}

// Strided max-pool over valid neighbors (dense C layout in and out).
__global__ void k_pool_max(const float* __restrict__ net, const int* __restrict__ idx,
                           const float* __restrict__ valid, int N, int C,
                           float* __restrict__ out) {
    const int b = blockIdx.y, mo = blockIdx.x;
    const int ms = mo * 2;
    __shared__ int   sI[KNN_K];
    __shared__ float sV[KNN_K];
    const size_t base = ((size_t)b * N + ms) * KNN_K;
    if (threadIdx.x < KNN_K) {
        sI[threadIdx.x] = idx[base + threadIdx.x];
        sV[threadIdx.x] = valid[base + threadIdx.x];
    }
    __syncthreads();
    for (int c = threadIdx.x; c < C; c += blockDim.x) {
        float mx = -3.4e38f;
        for (int k = 0; k < KNN_K; ++k) {
            if (sV[k] > 0.5f) mx = fmaxf(mx, net[((size_t)b * N + sI[k]) * C + c]);
        }
        out[((size_t)b * (N / 2) + mo) * C + c] = mx;
    }
}

__global__ void k_subsample_xyz(const float* __restrict__ xyz, int N,
                                float* __restrict__ out, int total) {
    const int t = blockIdx.x * blockDim.x + threadIdx.x;
    if (t >= total) return;
    const int d = t % 3, bm = t / 3;
    const int Nh = N / 2;
    const int b = bm / Nh, m = bm - b * Nh;
    out[t] = xyz[((size_t)b * N + 2 * m) * 3 + d];
}

// Global feature: max over points, scattered into the 832-wide concat buffer.
__global__ void k_feat_max(const float* __restrict__ net, int M, int C,
                           float* __restrict__ y, int ldy, int colOff, int total) {
    const int t = blockIdx.x * blockDim.x + threadIdx.x;
    if (t >= total) return;
    const int b = t / C, c = t - b * C;
    float mx = -3.4e38f;
    for (int m = 0; m < M; ++m) mx = fmaxf(mx, net[((size_t)b * M + m) * C + c]);
    y[(size_t)b * ldy + colOff + c] = mx;
}

// logits (padded 64-col) + bias -> d_out (B x 40).
__global__ void k_out_bias(const float* __restrict__ X, int ldx,
                           const float* __restrict__ bias,
                           float* __restrict__ out, int N, int total) {
    const int t = blockIdx.x * blockDim.x + threadIdx.x;
    if (t >= total) return;
    const int r = t / N, c = t - r * N;
    out[t] = X[(size_t)r * ldx + c] + bias[c];
}

// ---------------------------------------------------------------------------
// Host-side orchestration.
// ---------------------------------------------------------------------------
static inline size_t align256(size_t x) { return (x + 255) & ~(size_t)255; }

extern "C" void kernel_launch(void* const* d_in, const int* in_sizes, int n_in,
                              void* d_out, int out_size, void* d_ws, size_t ws_size,
                              hipStream_t stream) {
    (void)in_sizes; (void)n_in; (void)out_size; (void)ws_size;
    const int B = 4;
    const float RADIUS[3] = {0.2f, 0.4f, 0.8f};
    const int C_OUT[6] = {64, 64, 128, 128, 128, 128};
    const int colOff[3] = {0, 64, 192};

    // ---- inputs (setup_inputs dict order, flattened) ----
    const float* pc   = (const float*)d_in[0];
    const float* fc1w = (const float*)d_in[1];
    const float* fc1g = (const float*)d_in[2];
    const float* fc1b = (const float*)d_in[3];
    struct Conv { const float *depth, *point, *gamma, *beta; };
    Conv cv[6];
    for (int i = 0; i < 6; ++i) {
        cv[i].depth = (const float*)d_in[4 + 4 * i];
        cv[i].point = (const float*)d_in[5 + 4 * i];
        cv[i].gamma = (const float*)d_in[6 + 4 * i];
        cv[i].beta  = (const float*)d_in[7 + 4 * i];
    }
    const float* gdepth = (const float*)d_in[28];
    const float* gpoint = (const float*)d_in[29];
    const float* ggamma = (const float*)d_in[30];
    const float* gbeta  = (const float*)d_in[31];
    const float* w1 = (const float*)d_in[32];
    const float* g1 = (const float*)d_in[33];
    const float* b1 = (const float*)d_in[34];
    const float* w2 = (const float*)d_in[35];
    const float* g2 = (const float*)d_in[36];
    const float* b2 = (const float*)d_in[37];
    const float* w3 = (const float*)d_in[38];
    const float* b3 = (const float*)d_in[39];
    float* outp = (float*)d_out;

    // ---- workspace bump allocator ----
    char* wsp = (char*)d_ws;
    auto alloc = [&](size_t bytes) -> void* {
        void* p = (void*)wsp; wsp += align256(bytes); return p;
    };
    float* xyzL[4];
    xyzL[0] = (float*)alloc((size_t)B * 4096 * 3 * 4);
    xyzL[1] = (float*)alloc((size_t)B * 2048 * 3 * 4);
    xyzL[2] = (float*)alloc((size_t)B * 1024 * 3 * 4);
    xyzL[3] = (float*)alloc((size_t)B * 512  * 3 * 4);
    float* xyzPad = (float*)alloc((size_t)B * 4096 * 16 * 4);
    float* query  = (float*)alloc((size_t)B * 3 * 4);
    const size_t bigBytes = (size_t)B * 4096 * 144 * 4;
    float* bX = (float*)alloc(bigBytes);   // persistent "net"
    float* bC = (float*)alloc(bigBytes);   // concat / second pw output
    float* bD = (float*)alloc(bigBytes);   // depthwise output (padded K)
    int*   idxb  = (int*)  alloc((size_t)B * 4096 * KNN_K * 4);
    float* dstb  = (float*)alloc((size_t)B * 4096 * KNN_K * 4);
    float* valb  = (float*)alloc((size_t)B * 4096 * KNN_K * 4);
    int*   filtb = (int*)  alloc((size_t)B * 4096 * KNN_K * 4);
    float* statM = (float*)alloc(512 * 4);
    float* statI = (float*)alloc(512 * 4);
    // classifier-side buffers: 64 padded rows each
    float* ybuf   = (float*)alloc((size_t)64 * 832 * 4);
    float* h1     = (float*)alloc((size_t)64 * 512 * 4);
    float* h2     = (float*)alloc((size_t)64 * 256 * 4);
    float* gd     = (float*)alloc((size_t)64 * 128 * 4);
    float* gout   = (float*)alloc((size_t)64 * 512 * 4);
    float* logits = (float*)alloc((size_t)64 * 64 * 4);
    int*   gfilt  = (int*)  alloc((size_t)B * 512 * 4);
    // padded weight copies (K x Np, K already padded where needed)
    auto walloc = [&](int Kp, int Np) { return (float*)alloc((size_t)Kp * Np * 4); };
    float* fc1wp = walloc(16, 64);
    float* pw[6];
    const int pwKp[6] = {48, 64, 80, 128, 144, 128};
    const int pwK [6] = {35, 64, 67, 128, 131, 128};
    const int pwNp[6] = {64, 64, 128, 128, 128, 128};
    for (int i = 0; i < 6; ++i) pw[i] = walloc(pwKp[i], pwNp[i]);
    float* gpointp = walloc(128, 512);
    float* w1p = walloc(832, 512);
    float* w2p = walloc(512, 256);
    float* w3p = walloc(256, 64);

    auto gemm = [&](const float* A, const float* Bm, float* C,
                    int M, int N, int K, int lda, int ldb, int ldc) {
        dim3 grid(N / GEMM_NT, M / GEMM_MT);      // exact: all dims padded
        k_gemm_wmma<<<grid, 256, 0, stream>>>(A, Bm, C, M, N, K, lda, ldb, ldc);
    };
    auto bn = [&](float* X, int M, int N, int ldx, const float* g, const float* be,
                  float* out, int ldo, int co, int relu) {
        k_colstats<<<N, 256, 0, stream>>>(X, M, N, ldx, statM, statI);
        const int tot = M * N;
        k_bn_act<<<cdiv(tot, 256), 256, 0, stream>>>(X, M, N, ldx, statM, statI,
                                                     g, be, out, ldo, co, relu);
    };
    auto padw = [&](const float* src, int K, int N, float* dst, int Kp, int Np) {
        const int tot = Kp * Np;
        k_pad_copy<<<cdiv(tot, 256), 256, 0, stream>>>(src, K, N, dst, Np, tot);
    };
    auto zero = [&](float* p, int total) {
        k_zero<<<cdiv(total, 256), 256, 0, stream>>>(p, total);
    };

    // ---- per-call zero fills (pad regions) + weight pad-copies ----
    zero(xyzPad, B * 4096 * 16);
    zero(ybuf, 64 * 832);
    zero(gd, 64 * 128);
    padw(fc1w, 3, 32, fc1wp, 16, 64);
    for (int i = 0; i < 6; ++i) padw(cv[i].point, pwK[i], C_OUT[i], pw[i], pwKp[i], pwNp[i]);
    padw(gpoint, 128, 512, gpointp, 128, 512);
    padw(w1, 832, 512, w1p, 832, 512);
    padw(w2, 512, 256, w2p, 512, 256);
    padw(w3, 256, 40, w3p, 256, 64);

    // ---- normalize + query ----
    k_center_scale<<<B, 256, 0, stream>>>(pc, xyzL[0], xyzPad, query, 4096);

    // ---- fc1: (B*4096,16)@(16,64) -> BN -> ReLU (real shape (.,3)@(3,32)) ----
    gemm(xyzPad, fc1wp, bX, B * 4096, 64, 16, 16, 64, 64);
    bn(bX, B * 4096, 32, 64, fc1g, fc1b, bX, 64, 0, 1);

    int curC = 32, curLd = 64;
    int N = 4096;
    for (int l = 0; l < 3; ++l) {
        const int Cin1  = curC + 3;            // after concat (35/67/131)
        const int Cp1   = pwKp[2 * l];         // padded (48/80/144)
        const int Cout1 = C_OUT[2 * l];
        const int Cout2 = C_OUT[2 * l + 1];

        // concat(net, xyz) -> bC (zero-padded to Cp1)
        {
            const int tot = B * N * Cp1;
            k_concat<<<cdiv(tot, 256), 256, 0, stream>>>(bX, curLd, xyzL[l], N, curC,
                                                         bC, Cp1, tot);
        }
        // graph + bins
        {
            dim3 grid(cdiv(N, 64), B);
            k_build_graph<<<grid, 64, 0, stream>>>(xyzL[l], N, RADIUS[l], idxb, dstb, valb);
            const int tot = B * N * KNN_K;
            k_sph_filt<<<cdiv(tot, 256), 256, 0, stream>>>(xyzL[l], idxb, dstb, N,
                                                           RADIUS[l], filtb, tot);
        }
        // sep_conv #1: bC -> dw bD (K=Cp1) -> pw bX -> BN/ReLU in place
        {
            dim3 grid(N, B);
            k_depthwise<<<grid, 160, 0, stream>>>(bC, Cp1, idxb, filtb, valb,
                                                  cv[2 * l].depth, N, Cin1, Cp1, bD);
            gemm(bD, pw[2 * l], bX, B * N, Cout1, Cp1, Cp1, Cout1, Cout1);
            bn(bX, B * N, Cout1, Cout1, cv[2 * l].gamma, cv[2 * l].beta, bX, Cout1, 0, 1);
        }
        // sep_conv #2: bX -> dw bD (K=Cout1) -> pw bC -> BN/ReLU in place
        {
            dim3 grid(N, B);
            k_depthwise<<<grid, 160, 0, stream>>>(bX, Cout1, idxb, filtb, valb,
                                                  cv[2 * l + 1].depth, N, Cout1, Cout1, bD);
            gemm(bD, pw[2 * l + 1], bC, B * N, Cout2, Cout1, Cout1, Cout2, Cout2);
            bn(bC, B * N, Cout2, Cout2, cv[2 * l + 1].gamma, cv[2 * l + 1].beta,
               bC, Cout2, 0, 1);
        }
        // pool (stride 2) bC -> bX; subsample xyz; global feature max
        {
            dim3 grid(N / 2, B);
            k_pool_max<<<grid, 160, 0, stream>>>(bC, idxb, valb, N, Cout2, bX);
            const int tot = B * (N / 2) * 3;
            k_subsample_xyz<<<cdiv(tot, 256), 256, 0, stream>>>(xyzL[l], N, xyzL[l + 1], tot);
            const int tf = B * Cout2;
            k_feat_max<<<cdiv(tf, 256), 256, 0, stream>>>(bX, N / 2, Cout2, ybuf, 832,
                                                          colOff[l], tf);
        }
        curC = Cout2; curLd = Cout2;
        N /= 2;
    }

    // ---- global conv over all 512 remaining points, query center ----
    {
        const int tot = B * 512;
        k_global_filt<<<cdiv(tot, 256), 256, 0, stream>>>(xyzL[3], query, 512, gfilt, tot);
        k_global_dw<<<B, 128, 0, stream>>>(bX, gfilt, gdepth, 512, 128, gd);
        gemm(gd, gpointp, gout, 64, 512, 128, 128, 512, 512);       // rows 4..63 zero
        bn(gout, B, 512, 512, ggamma, gbeta, ybuf, 832, 320, 1);    // y[:,320:832]
    }

    // ---- classifier head (M padded to 64; pad rows stay zero) ----
    gemm(ybuf, w1p, h1, 64, 512, 832, 832, 512, 512);
    bn(h1, B, 512, 512, g1, b1, h1, 512, 0, 1);
    gemm(h1, w2p, h2, 64, 256, 512, 512, 256, 256);
    bn(h2, B, 256, 256, g2, b2, h2, 256, 0, 1);
    gemm(h2, w3p, logits, 64, 64, 256, 256, 64, 64);
    k_out_bias<<<1, 256, 0, stream>>>(logits, 64, b3, outp, 40, B * 40);
}